// HSVLoss_37168646979883
// MI455X (gfx1250) — compile-verified
//
#include <hip/hip_runtime.h>

// ---------------------------------------------------------------------------
// HSV hue loss, MI455X (gfx1250, wave32).
// Bandwidth-bound: 201 MB streamed once -> ~8.6 us floor at 23.3 TB/s.
// Hot loop: 6x global_load_b128 (non-temporal, immediate plane offsets) per
// iteration; hue math is fully branchless (v_cndmask, v_rcp_f32) so VALU time
// stays under the HBM floor. Cross-lane reduction: V_WMMA_F32_16X16X4_F32.
// ---------------------------------------------------------------------------

typedef float vf4 __attribute__((ext_vector_type(4)));
typedef float v2f __attribute__((ext_vector_type(2)));
typedef float v8f __attribute__((ext_vector_type(8)));

#define HW     262144   // 512*512 = 2^18
#define NPIX   8388608  // 32*512*512 = 2^23
#define NQUAD  2097152  // NPIX/4
#define NBLK   2048
#define NTHR   256      // 8 wave32s per block

// Branchless; matches jnp.where nesting ((maxc==r) first, then (maxc==g)),
// h = (h/6) mod 1 = x - floor(x); h forced to 0 where delta==0.
__device__ __forceinline__ float rgb_to_hue(float r, float g, float b) {
    float mx  = fmaxf(r, fmaxf(g, b));       // v_max3_num_f32
    float mn  = fminf(r, fminf(g, b));       // v_min3_num_f32
    float d   = mx - mn;
    bool  dz  = (d == 0.0f);
    float inv = __builtin_amdgcn_rcpf(dz ? 1.0f : d);   // v_rcp_f32, 1 ulp
    float hr  = (g - b) * inv;               // maxc == r candidate
    float hg  = fmaf(b - r, inv, 2.0f);      // maxc == g candidate
    float hb  = fmaf(r - g, inv, 4.0f);      // else candidate
    float h   = (mx == r) ? hr : ((mx == g) ? hg : hb);  // pure-value selects
    h *= 0.166666666666666666f;
    h -= floorf(h);
    return dz ? 0.0f : h;
}

// diff*[diff<0.5] + (diff-0.5)*[diff>0.5]  ==  diff - (diff>=0.5 ? 0.5 : 0)
// (both masks are zero at diff==0.5, and this form gives exactly 0 there).
__device__ __forceinline__ float hue_term(float hp, float ht) {
    float diff = fabsf(hp - ht);
    return diff - ((diff >= 0.5f) ? 0.5f : 0.0f);
}

// Wave32 all-lane sum via the matrix pipe (exact f32):
// A(16x4): lane L<16 holds {acc_L, 0} (K=0,1); lane L>=16 holds {acc_L, 0} (K=2,3)
//   => A[m][0]=acc_m, A[m][2]=acc_{m+16}, odd-K columns zero.
// B(4x16) = ones => D[m][n] = acc_m + acc_{m+16} for every n.
// Per-lane sum of D's 8 VGPRs gives half-sums; xor-16 shuffle merges halves.
__device__ __forceinline__ float wave_sum_wmma(float acc) {
    v2f a; a.x = acc;  a.y = 0.0f;
    v2f b; b.x = 1.0f; b.y = 1.0f;
    v8f c = {};
    v8f d = __builtin_amdgcn_wmma_f32_16x16x4_f32(
        /*neg_a=*/false, a, /*neg_b=*/false, b,
        /*c_mod=*/(short)0, c, /*reuse_a=*/false, /*reuse_b=*/false);
    float s = ((d[0] + d[1]) + (d[2] + d[3])) + ((d[4] + d[5]) + (d[6] + d[7]));
    s += __shfl_xor(s, 16, 32);
    return s;   // full 32-lane sum, replicated across the wave
}

__device__ __forceinline__ void block_reduce_store(float acc, float* dst, float scale) {
    __shared__ float swave[NTHR / 32];
    float ws = wave_sum_wmma(acc);
    const int lane = threadIdx.x & 31;
    const int wid  = threadIdx.x >> 5;
    if (lane == 0) swave[wid] = ws;
    __syncthreads();
    if (threadIdx.x == 0) {
        float t = 0.0f;
        #pragma unroll
        for (int i = 0; i < NTHR / 32; ++i) t += swave[i];
        *dst = t * scale;
    }
}

__global__ __launch_bounds__(NTHR) void hue_loss_partial(
    const float* __restrict__ pred, const float* __restrict__ targ,
    float* __restrict__ partial)
{
    const int tid = blockIdx.x * NTHR + threadIdx.x;
    float acc = 0.0f;

    // Quad q -> image b = q>>16 (HW/4 = 2^16), in-plane offset (q&65535)*4.
    // Planes R/G/B are contiguous: base, base+HW, base+2*HW (imm offsets).
    for (int q = tid; q < NQUAD; q += NBLK * NTHR) {
        const int b    = q >> 16;
        const int base = b * (3 * HW) + ((q & 65535) << 2);

        vf4 pR = __builtin_nontemporal_load((const vf4*)(pred + base));
        vf4 pG = __builtin_nontemporal_load((const vf4*)(pred + base + HW));
        vf4 pB = __builtin_nontemporal_load((const vf4*)(pred + base + 2 * HW));
        vf4 tR = __builtin_nontemporal_load((const vf4*)(targ + base));
        vf4 tG = __builtin_nontemporal_load((const vf4*)(targ + base + HW));
        vf4 tB = __builtin_nontemporal_load((const vf4*)(targ + base + 2 * HW));

        acc += hue_term(rgb_to_hue(pR.x, pG.x, pB.x), rgb_to_hue(tR.x, tG.x, tB.x));
        acc += hue_term(rgb_to_hue(pR.y, pG.y, pB.y), rgb_to_hue(tR.y, tG.y, tB.y));
        acc += hue_term(rgb_to_hue(pR.z, pG.z, pB.z), rgb_to_hue(tR.z, tG.z, tB.z));
        acc += hue_term(rgb_to_hue(pR.w, pG.w, pB.w), rgb_to_hue(tR.w, tG.w, tB.w));
    }

    block_reduce_store(acc, &partial[blockIdx.x], 1.0f);
}

__global__ __launch_bounds__(NTHR) void hue_loss_final(
    const float* __restrict__ partial, float* __restrict__ out)
{
    float acc = 0.0f;
    for (int i = threadIdx.x; i < NBLK; i += NTHR) acc += partial[i];
    // N = 2^23 -> exact scale by 0x1p-23f.
    block_reduce_store(acc, out, 0x1p-23f);
}

extern "C" void kernel_launch(void* const* d_in, const int* in_sizes, int n_in,
                              void* d_out, int out_size, void* d_ws, size_t ws_size,
                              hipStream_t stream) {
    const float* pred = (const float*)d_in[0];
    const float* targ = (const float*)d_in[1];
    float* partial    = (float*)d_ws;          // NBLK floats of scratch
    float* out        = (float*)d_out;

    hue_loss_partial<<<NBLK, NTHR, 0, stream>>>(pred, targ, partial);
    hue_loss_final<<<1, NTHR, 0, stream>>>(partial, out);
}